// CapsuleLayer_14671608283365
// MI455X (gfx1250) — compile-verified
//
#include <hip/hip_runtime.h>

typedef float v2f __attribute__((ext_vector_type(2)));
typedef float v8f __attribute__((ext_vector_type(8)));

#define BSZ   32
#define IN_C  128
#define DIN   256
#define CH    32
#define RP    4          // IN_C / CH
#define NC    47
#define DC    64
#define NCD   (NC * DC)  // 3008

#define KC     32
#define NCHUNK (DIN / KC)   // 8
#define LDSTR  36        // padded LDS row stride (floats): 16B aligned, conflict-free b64 frag reads
#define HSTR   68        // routing LDS row stride (floats): 16B aligned

// ---- CDNA5 async global->LDS copy (ASYNCcnt path), per CDNA5 ISA ----
__device__ __forceinline__ unsigned lds_off32(const float* p) {
  // AS3 pointer's integer value is the group-segment (LDS) offset
  return (unsigned)(uintptr_t)(__attribute__((address_space(3))) const float*)p;
}

__device__ __forceinline__ void async_copy16(const float* lds_dst, const float* gsrc) {
  unsigned loff = lds_off32(lds_dst);
  unsigned long long ga = (unsigned long long)(uintptr_t)gsrc;
  asm volatile("global_load_async_to_lds_b128 %0, %1, off"
               :: "v"(loff), "v"(ga)
               : "memory");
}

__device__ __forceinline__ void wait_async0() {
  asm volatile("s_wait_asynccnt 0" ::: "memory");
}

// ---------------------------------------------------------------------------
// Phase 1: inputs_hat[b, c*4+rr, n*64+k] = sum_d x[b,c,rr,d] * W[c,n,k,d]
// Per channel: GEMM M=128 (b,rr), N=3008 (n,k), K=256 via v_wmma_f32_16x16x4_f32.
// Block: 256 threads (8 waves). Block tile M=128 x N=64; wave tile M=16 x N=64.
// Double-buffered LDS fed by global_load_async_to_lds_b128 (ASYNCcnt).
// ---------------------------------------------------------------------------
__global__ __launch_bounds__(256) void caps_gemm_wmma(
    const float* __restrict__ inp,   // [32,128,256]
    const float* __restrict__ W,     // [32,47,64,256]
    float* __restrict__ hat)         // [32,128,3008]
{
  __shared__ float Alds[2][128 * LDSTR];   // 2 x 18 KB
  __shared__ float Blds[2][64 * LDSTR];    // 2 x  9 KB

  const int nb   = blockIdx.x;          // 0..46  (64-column N block)
  const int c    = blockIdx.y;          // 0..31  (channel)
  const int t    = threadIdx.x;         // 0..255
  const int lane = t & 31;
  const int w    = t >> 5;              // wave id -> M tile (16 rows)
  const int nlane = lane & 15;
  const int khalf = (lane >> 4) << 1;   // 0 or 2 (K-pair select per ISA A/B layout)
  const int mrow  = (w << 4) + nlane;   // A row this lane feeds

  const float* inp_c = inp + (size_t)c * RP * DIN;
  const float* Wc    = W + (size_t)c * NC * DC * DIN + (size_t)nb * 64 * DIN;

  // precompute this thread's staging coordinates (same every chunk)
  int am[4], aj[4], brow[2], bj[2];
  const float* ag[4];
  const float* bg[2];
  #pragma unroll
  for (int q = 0; q < 4; ++q) {
    int idx = t + q * 256;              // 0..1023 float4s of the A tile
    am[q] = idx >> 3;                   // row m: b = m>>2, rr = m&3
    aj[q] = idx & 7;                    // float4 slot within 32-float row
    int bi = am[q] >> 2, rr = am[q] & 3;
    ag[q] = inp_c + (size_t)(bi * IN_C + rr) * DIN + (aj[q] << 2);
  }
  #pragma unroll
  for (int q = 0; q < 2; ++q) {
    int idx = t + q * 256;              // 0..511 float4s of the B tile
    brow[q] = idx >> 3;
    bj[q]   = idx & 7;
    bg[q] = Wc + (size_t)brow[q] * DIN + (bj[q] << 2);
  }

  v8f acc[4] = {v8f{}, v8f{}, v8f{}, v8f{}};

  // prologue: issue chunk 0 into buffer 0
  #pragma unroll
  for (int q = 0; q < 4; ++q)
    async_copy16(&Alds[0][am[q] * LDSTR + (aj[q] << 2)], ag[q]);
  #pragma unroll
  for (int q = 0; q < 2; ++q)
    async_copy16(&Blds[0][brow[q] * LDSTR + (bj[q] << 2)], bg[q]);

  for (int chunk = 0; chunk < NCHUNK; ++chunk) {
    const int cur = chunk & 1;
    wait_async0();          // this wave's copies into buf[cur] are done
    __syncthreads();        // => everyone's copies landed; everyone done reading buf[cur^1]

    if (chunk + 1 < NCHUNK) {
      const int k1 = (chunk + 1) * KC;
      #pragma unroll
      for (int q = 0; q < 4; ++q)
        async_copy16(&Alds[cur ^ 1][am[q] * LDSTR + (aj[q] << 2)], ag[q] + k1);
      #pragma unroll
      for (int q = 0; q < 2; ++q)
        async_copy16(&Blds[cur ^ 1][brow[q] * LDSTR + (bj[q] << 2)], bg[q] + k1);
    }

    const float* Ab = &Alds[cur][0];
    const float* Bb = &Blds[cur][0];
    #pragma unroll
    for (int kk = 0; kk < KC; kk += 4) {
      const v2f a = *(const v2f*)&Ab[mrow * LDSTR + kk + khalf];
      #pragma unroll
      for (int tt = 0; tt < 4; ++tt) {
        const v2f b = *(const v2f*)&Bb[((tt << 4) + nlane) * LDSTR + kk + khalf];
        acc[tt] = __builtin_amdgcn_wmma_f32_16x16x4_f32(
            false, a, false, b, (short)0, acc[tt], false, false);
      }
    }
  }

  // ---- epilogue: C/D layout -> global (lanes contiguous in n => 64B segments) ----
  const int mofs = (lane >> 4) << 3;    // 0 or 8
  #pragma unroll
  for (int tt = 0; tt < 4; ++tt) {
    #pragma unroll
    for (int j = 0; j < 8; ++j) {
      int m  = (w << 4) + mofs + j;
      int bi = m >> 2, rr = m & 3;
      int col = nb * 64 + (tt << 4) + nlane;
      hat[(size_t)(bi * IN_C + c * RP + rr) * NCD + col] = acc[tt][j];
    }
  }
}

// ---------------------------------------------------------------------------
// Phase 2: fused dynamic routing. One block per (b, n); hat slice [128 x 64]
// staged into LDS once, all 3 routing iterations done locally.
// ---------------------------------------------------------------------------
__global__ __launch_bounds__(256) void caps_routing_fused(
    const float* __restrict__ hat,   // [32,128,3008]
    const float* __restrict__ Bp,    // [47,64]
    float* __restrict__ out)         // [32,47,64]
{
  __shared__ float h[IN_C * HSTR];   // 34.8 KB (padded rows)
  __shared__ float blog[IN_C];
  __shared__ float cc[IN_C];
  __shared__ float sv[DC];
  __shared__ float ov[DC];
  __shared__ float red[IN_C];

  const int n = blockIdx.x;          // 0..46
  const int b = blockIdx.y;          // 0..31
  const int t = threadIdx.x;         // 0..255

  // ---- stage hat[b, :, n, :] : 128 x 64 floats, float4 coalesced ----
  const float* src = hat + (size_t)b * IN_C * NCD + n * DC;
  #pragma unroll
  for (int q = 0; q < 8; ++q) {
    int idx = t + q * 256;           // 0..2047 float4s
    int i   = idx >> 4;              // input capsule
    int j   = idx & 15;              // float4 slot
    float4 v = *(const float4*)(src + (size_t)i * NCD + (j << 2));
    *(float4*)&h[i * HSTR + (j << 2)] = v;
  }
  if (t < IN_C) blog[t] = 0.0f;
  __syncthreads();

  for (int it = 0; it < 3; ++it) {
    // ---- softmax over the 128 input capsules ----
    if (t < IN_C) red[t] = blog[t];
    __syncthreads();
    for (int off = 64; off > 0; off >>= 1) {
      if (t < off) red[t] = fmaxf(red[t], red[t + off]);
      __syncthreads();
    }
    float mx = red[0];
    __syncthreads();
    if (t < IN_C) { float e = expf(blog[t] - mx); cc[t] = e; red[t] = e; }
    __syncthreads();
    for (int off = 64; off > 0; off >>= 1) {
      if (t < off) red[t] += red[t + off];
      __syncthreads();
    }
    float inv = 1.0f / red[0];
    __syncthreads();
    if (t < IN_C) cc[t] *= inv;
    __syncthreads();

    // ---- s[k] = sum_i c[i]*hat[i,k] + B_param[n,k] ; then squash ----
    if (t < DC) {
      float s = Bp[n * DC + t];
      for (int i = 0; i < IN_C; ++i) s = fmaf(cc[i], h[i * HSTR + t], s);
      sv[t]  = s;
      red[t] = s * s;
    }
    __syncthreads();
    for (int off = 32; off > 0; off >>= 1) {
      if (t < off) red[t] += red[t + off];
      __syncthreads();
    }
    float s2    = red[0];
    float scale = s2 / (1.0f + s2) * rsqrtf(s2 + 1e-7f);
    if (t < DC) ov[t] = scale * sv[t];
    __syncthreads();

    // ---- agreement: blog[i] += <out, hat[i,:]> ----
    if (it < 2) {
      if (t < IN_C) {
        float d = 0.0f;
        for (int k = 0; k < DC; ++k) d = fmaf(ov[k], h[t * HSTR + k], d);
        blog[t] += d;
      }
      __syncthreads();
    }
  }

  if (t < DC) out[((size_t)b * NC + n) * DC + t] = ov[t];
}

// ---------------------------------------------------------------------------
extern "C" void kernel_launch(void* const* d_in, const int* in_sizes, int n_in,
                              void* d_out, int out_size, void* d_ws, size_t ws_size,
                              hipStream_t stream) {
  const float* inp = (const float*)d_in[0];   // [32,128,256]
  const float* W   = (const float*)d_in[1];   // [32,47,64,256]
  const float* Bp  = (const float*)d_in[2];   // [47,64]
  float* out = (float*)d_out;                 // [32,47,64]
  float* hat = (float*)d_ws;                  // needs 32*128*3008*4 = 49.3 MB

  caps_gemm_wmma<<<dim3(NC, CH), 256, 0, stream>>>(inp, W, hat);
  caps_routing_fused<<<dim3(NC, BSZ), 256, 0, stream>>>(hat, Bp, out);
}